// LOSS_69312182222886
// MI455X (gfx1250) — compile-verified
//
#include <hip/hip_runtime.h>
#include <hip/hip_bf16.h>
#include <math.h>

typedef float v2f __attribute__((ext_vector_type(2)));
typedef float v8f __attribute__((ext_vector_type(8)));

#define NANC   17064            // total anchors across 5 levels
#define NANC16 17072            // padded to multiple of 16 for WMMA tiles
#define NB     16               // batch
#define NG     32               // GT boxes per image
#define NCLS   81
#define BGLBL  81
#define BN     (NB * NANC)      // 273024 samples
#define EPSF   1e-6f
#define LDSW   17               // padded LDS row stride (floats) to avoid bank conflicts

// ---------------------------------------------------------------------------
// helpers
// ---------------------------------------------------------------------------
__device__ inline float wave_sum32(float v) {
#pragma unroll
    for (int off = 16; off > 0; off >>= 1) v += __shfl_down(v, off, 32);
    return v;
}

struct Loc { int lvl, b, pix, n, hw; };

__device__ inline Loc locate_sample(int s) {
    // sample-major order: level 0 block [0,204800), then level 1 ...
    const int LB[6] = {0, 204800, 256000, 268800, 272128, 273024};
    const int NP[5] = {12800, 3200, 800, 208, 56};
    const int OFF[5] = {0, 12800, 16000, 16800, 17008};
    int lvl = 0;
#pragma unroll
    for (int i = 1; i < 5; ++i) if (s >= LB[i]) lvl = i;
    Loc L;
    int r = s - LB[lvl];
    L.lvl = lvl; L.hw = NP[lvl];
    L.b = r / NP[lvl]; L.pix = r - L.b * NP[lvl];
    L.n = OFF[lvl] + L.pix;
    return L;
}

// ---------------------------------------------------------------------------
// K0: zero accumulators and scatter arrays
// ---------------------------------------------------------------------------
__global__ void k_init(float* __restrict__ acc, int* __restrict__ s12, int n) {
    int i = blockIdx.x * blockDim.x + threadIdx.x;
    if (i < 8) acc[i] = 0.0f;
    for (int j = i; j < n; j += gridDim.x * blockDim.x) s12[j] = 0;
}

// ---------------------------------------------------------------------------
// K1: anchor rank-8 features for rf1 / rf2 + point coordinates
// ---------------------------------------------------------------------------
__global__ void k_anchor(float* __restrict__ V1, float* __restrict__ V2,
                         float* __restrict__ P) {
    int n = blockIdx.x * blockDim.x + threadIdx.x;
    if (n >= NANC16) return;
    if (n >= NANC) {                 // pad rows: zero features (masked in scan)
#pragma unroll
        for (int k = 0; k < 8; ++k) { V1[n * 8 + k] = 0.0f; V2[n * 8 + k] = 0.0f; }
        return;
    }
    const int   OFF[6] = {0, 12800, 16000, 16800, 17008, 17064};
    const int   WID[5] = {128, 64, 32, 16, 8};
    const int   STR[5] = {8, 16, 32, 64, 128};
    const float RF[5]  = {45.5f, 133.5f, 213.5f, 277.5f, 405.5f}; // trf[l+1]*0.5
    int lvl = 0;
#pragma unroll
    for (int i = 1; i < 5; ++i) if (n >= OFF[i]) lvl = i;
    int pix = n - OFF[lvl];
    int row = pix / WID[lvl], col = pix - row * WID[lvl];
    float s = (float)STR[lvl];
    float x = col * s + 0.5f * s;
    float y = row * s + 0.5f * s;
    P[n * 2 + 0] = x; P[n * 2 + 1] = y;
    float rf = RF[lvl];
    {
        float ww = rf + EPSF; float w2s = ww * ww;
        float* v = V1 + n * 8;
        v[0] = w2s; v[1] = w2s; v[2] = 1.0f; v[3] = x; v[4] = x * x;
        v[5] = y;   v[6] = y * y; v[7] = -2.0f * logf(w2s);
    }
    {
        float ww = rf * 0.9f + EPSF; float w2s = ww * ww;
        float* v = V2 + n * 8;
        v[0] = w2s; v[1] = w2s; v[2] = 1.0f; v[3] = x; v[4] = x * x;
        v[5] = y;   v[6] = y * y; v[7] = -2.0f * logf(w2s);
    }
}

// ---------------------------------------------------------------------------
// K2: GT rank-8 features
// ---------------------------------------------------------------------------
__global__ void k_gtfeat(const float* __restrict__ gtb, float* __restrict__ U) {
    int i = blockIdx.x * blockDim.x + threadIdx.x;
    if (i >= NB * NG) return;
    const float* g = gtb + i * 4;
    float x1 = g[0], y1 = g[1], x2 = g[2], y2 = g[3];
    float c1x = 0.5f * (x1 + x2), c1y = 0.5f * (y1 + y2);
    float w1 = x2 - x1 + EPSF, h1 = y2 - y1 + EPSF;
    float w1s = w1 * w1, h1s = h1 * h1;
    float iw = 1.0f / w1s, ih = 1.0f / h1s;
    float* u = U + i * 8;
    u[0] = iw; u[1] = ih;
    u[2] = 4.0f * c1x * c1x * iw + 4.0f * c1y * c1y * ih + logf(w1s) + logf(h1s) - 2.0f;
    u[3] = -8.0f * c1x * iw; u[4] = 4.0f * iw;
    u[5] = -8.0f * c1y * ih; u[6] = 4.0f * ih; u[7] = 1.0f;
}

// ---------------------------------------------------------------------------
// K3: WMMA overlap tiles + per-GT top-3 (ov1) / top-1 (ov2) + scatter-max.
// One workgroup (1 wave32) per image. EXEC all-1s for WMMA.
// ov tile (32 GT x 16 anchors) = (32x8)@(8x16) fp32 GEMM via
// V_WMMA_F32_16X16X4_F32 (K=8 in two K=4 steps) per 16-GT row block.
// ov = 1/(1+kl) = 2/(2+dot) is monotone decreasing in dot, so top-k(ov)
// == bottom-k(dot): track smallest dots directly, no division needed.
// ---------------------------------------------------------------------------
__global__ __launch_bounds__(32)
void k_assign(const float* __restrict__ U, const float* __restrict__ V1,
              const float* __restrict__ V2, int* __restrict__ s1,
              int* __restrict__ s2) {
    const int b = blockIdx.x;
    const int lane = threadIdx.x;
    __shared__ float o1[32 * LDSW];
    __shared__ float o2[32 * LDSW];

    const float* Ub = U + b * NG * 8;
    const int m    = lane & 15;
    const int koff = (lane >> 4) * 2;   // lanes 0-15 hold K=kb+0/1, 16-31 hold K=kb+2/3

    // A fragments are loop-invariant: [rowblock][kstep]
    v2f a[2][2];
#pragma unroll
    for (int rb = 0; rb < 2; ++rb)
#pragma unroll
        for (int ks = 0; ks < 2; ++ks) {
            const float* up = Ub + (16 * rb + m) * 8 + ks * 4 + koff;
            a[rb][ks].x = up[0];
            a[rb][ks].y = up[1];
        }

    // per-lane (lane == GT index) running bottom-3 of dot (== top-3 of ov)
    float m0 = 1e30f, m1 = 1e30f, m2 = 1e30f;
    int   i0 = 0, i1 = 0, i2 = 0;
    float w0 = 1e30f; int j0 = 0;

    for (int t0 = 0; t0 < NANC16; t0 += 16) {
        const int ncol = t0 + (lane & 15);
        const float* p1 = V1 + ncol * 8 + koff;
        const float* p2 = V2 + ncol * 8 + koff;
        v2f b1a, b1b, b2a, b2b;
        b1a.x = p1[0]; b1a.y = p1[1]; b1b.x = p1[4]; b1b.y = p1[5];
        b2a.x = p2[0]; b2a.y = p2[1]; b2b.x = p2[4]; b2b.y = p2[5];

#pragma unroll
        for (int rb = 0; rb < 2; ++rb) {
            v8f c1 = {};
            c1 = __builtin_amdgcn_wmma_f32_16x16x4_f32(false, a[rb][0], false, b1a,
                                                       (short)0, c1, false, false);
            c1 = __builtin_amdgcn_wmma_f32_16x16x4_f32(false, a[rb][1], false, b1b,
                                                       (short)0, c1, false, false);
            v8f c2 = {};
            c2 = __builtin_amdgcn_wmma_f32_16x16x4_f32(false, a[rb][0], false, b2a,
                                                       (short)0, c2, false, false);
            c2 = __builtin_amdgcn_wmma_f32_16x16x4_f32(false, a[rb][1], false, b2b,
                                                       (short)0, c2, false, false);
            const int rbase = rb * 16 + ((lane >> 4) * 8);
            const int col = lane & 15;
#pragma unroll
            for (int j = 0; j < 8; ++j) {
                o1[(rbase + j) * LDSW + col] = c1[j];
                o2[(rbase + j) * LDSW + col] = c2[j];
            }
        }
        __syncthreads();

        const int g = lane;                       // this lane owns GT g
        const int cmax = (NANC - t0 < 16) ? (NANC - t0) : 16;   // tail mask hoisted
        for (int c = 0; c < cmax; ++c) {
            const int anc = t0 + c;
            float d1 = o1[g * LDSW + c];
            if (d1 < m0)      { m2 = m1; i2 = i1; m1 = m0; i1 = i0; m0 = d1; i0 = anc; }
            else if (d1 < m1) { m2 = m1; i2 = i1; m1 = d1; i1 = anc; }
            else if (d1 < m2) { m2 = d1; i2 = anc; }
            float d2 = o2[g * LDSW + c];
            if (d2 < w0) { w0 = d2; j0 = anc; }
        }
        __syncthreads();
    }

    const int gid = lane + 1;                     // gt ids are 1..G
    atomicMax(&s1[b * NANC + i0], gid);
    atomicMax(&s1[b * NANC + i1], gid);
    atomicMax(&s1[b * NANC + i2], gid);
    atomicMax(&s2[b * NANC + j0], gid);
}

// ---------------------------------------------------------------------------
// K4: focal classification loss (sum of bce*fw over all samples/classes).
// Each thread owns 4 consecutive pixels (all level/batch block boundaries are
// multiples of 4), so the dominant 88MB logit stream uses b128 loads.
// ---------------------------------------------------------------------------
__global__ void k_cls(const float* __restrict__ c0, const float* __restrict__ c1,
                      const float* __restrict__ c2, const float* __restrict__ c3,
                      const float* __restrict__ c4, const int* __restrict__ s1,
                      const int* __restrict__ s2, const int* __restrict__ glab,
                      float* __restrict__ acc) {
    int t = blockIdx.x * blockDim.x + threadIdx.x;
    float lsum = 0.0f;
    if (t < BN / 4) {
        int s = t * 4;
        Loc L = locate_sample(s);                 // 4 samples share b, lvl
        const float* cp = (L.lvl == 0) ? c0 : (L.lvl == 1) ? c1 : (L.lvl == 2) ? c2
                          : (L.lvl == 3) ? c3 : c4;
        int lab[4];
#pragma unroll
        for (int j = 0; j < 4; ++j) {
            const int bn = L.b * NANC + L.n + j;
            int a2 = s2[bn], a1 = s1[bn];
            int a  = (a2 > 0) ? a2 : ((a1 > 0) ? a1 : 0);
            lab[j] = (a > 0) ? glab[L.b * NG + a - 1] : BGLBL;
        }
        const float* base = cp + (size_t)L.b * NCLS * L.hw + L.pix;
        for (int c = 0; c < NCLS; ++c) {
            float4 xv = *(const float4*)(base + c * L.hw);
            float xs[4] = {xv.x, xv.y, xv.z, xv.w};
#pragma unroll
            for (int j = 0; j < 4; ++j) {
                float x = xs[j];
                float tt = (c == lab[j]) ? 1.0f : 0.0f;
                float p = 1.0f / (1.0f + expf(-x));
                float pt = (1.0f - p) * tt + p * (1.0f - tt);
                float fw = (0.25f * tt + 0.75f * (1.0f - tt)) * pt * pt;
                float bce = fmaxf(x, 0.0f) - x * tt + log1pf(expf(-fabsf(x)));
                lsum += bce * fw;
            }
        }
    }
    lsum = wave_sum32(lsum);
    if ((threadIdx.x & 31) == 0) atomicAdd(&acc[0], lsum);
}

// ---------------------------------------------------------------------------
// K5: DIoU*centerness, centerness-target sum, centerness BCE, pos count
// ---------------------------------------------------------------------------
__global__ void k_box(const float* __restrict__ b0, const float* __restrict__ b1,
                      const float* __restrict__ b2, const float* __restrict__ b3,
                      const float* __restrict__ b4, const float* __restrict__ o0,
                      const float* __restrict__ o1, const float* __restrict__ o2,
                      const float* __restrict__ o3, const float* __restrict__ o4,
                      const float* __restrict__ gtb, const float* __restrict__ P,
                      const int* __restrict__ s1, const int* __restrict__ s2,
                      float* __restrict__ acc) {
    int s = blockIdx.x * blockDim.x + threadIdx.x;
    float csum = 0.0f, bsum = 0.0f, cbce = 0.0f, npos = 0.0f;
    if (s < BN) {
        Loc L = locate_sample(s);
        const int bn = L.b * NANC + L.n;
        int a2 = s2[bn], a1 = s1[bn];
        int a  = (a2 > 0) ? a2 : ((a1 > 0) ? a1 : 0);
        if (a > 0) {                               // positive sample
            const float* bp = (L.lvl == 0) ? b0 : (L.lvl == 1) ? b1 : (L.lvl == 2) ? b2
                              : (L.lvl == 3) ? b3 : b4;
            const float* op = (L.lvl == 0) ? o0 : (L.lvl == 1) ? o1 : (L.lvl == 2) ? o2
                              : (L.lvl == 3) ? o3 : o4;
            const float* g = gtb + (L.b * NG + a - 1) * 4;
            float x = P[L.n * 2 + 0], y = P[L.n * 2 + 1];
            float bl = x - g[0], bt = y - g[1], br = g[2] - x, bb = g[3] - y;
            float lrmin = fminf(bl, br), lrmax = fmaxf(bl, br);
            float tbmin = fminf(bt, bb), tbmax = fmaxf(bt, bb);
            float ctr = sqrtf(fmaxf((lrmin / lrmax) * (tbmin / tbmax), 0.0f));
            // predicted box
            const float* fb = bp + (size_t)L.b * 4 * L.hw + L.pix;
            float p0 = x - fb[0 * L.hw], p1 = y - fb[1 * L.hw];
            float p2 = x + fb[2 * L.hw], p3 = y + fb[3 * L.hw];
            float t0 = g[0], t1 = g[1], t2 = g[2], t3 = g[3];
            // DIoU
            float ltx = fmaxf(p0, t0), lty = fmaxf(p1, t1);
            float rbx = fminf(p2, t2), rby = fminf(p3, t3);
            float wx = fmaxf(rbx - ltx, 0.0f), wy = fmaxf(rby - lty, 0.0f);
            float ov = wx * wy;
            float ap = (p2 - p0) * (p3 - p1);
            float ag = (t2 - t0) * (t3 - t1);
            float iou = ov / (ap + ag - ov + EPSF);
            float ex = fmaxf(p2, t2) - fminf(p0, t0);
            float ey = fmaxf(p3, t3) - fminf(p1, t1);
            ex = fmaxf(ex, 0.0f); ey = fmaxf(ey, 0.0f);
            float cc = ex * ex + ey * ey + EPSF;
            float dx = (p0 + p2 - t0 - t2), dy = (p1 + p3 - t1 - t3);
            float rho2 = (dx * dx + dy * dy) * 0.25f;
            float diou = 1.0f - (iou - rho2 / cc);
            // centerness logit
            float fo = op[(size_t)L.b * L.hw + L.pix];
            cbce = fmaxf(fo, 0.0f) - fo * ctr + log1pf(expf(-fabsf(fo)));
            csum = ctr;
            bsum = diou * ctr;
            npos = 1.0f;
        }
    }
    bsum = wave_sum32(bsum);
    csum = wave_sum32(csum);
    npos = wave_sum32(npos);
    cbce = wave_sum32(cbce);
    if ((threadIdx.x & 31) == 0) {
        atomicAdd(&acc[1], bsum);
        atomicAdd(&acc[2], csum);
        atomicAdd(&acc[3], npos);
        atomicAdd(&acc[4], cbce);
    }
}

// ---------------------------------------------------------------------------
// K6: finalize the three scalars
// ---------------------------------------------------------------------------
__global__ void k_final(const float* __restrict__ acc, float* __restrict__ out) {
    float np = fmaxf(acc[3], 1.0f);
    out[0] = acc[0] / np;
    out[1] = acc[1] / fmaxf(acc[2], 1e-6f);
    out[2] = acc[4] / np;
}

// ---------------------------------------------------------------------------
extern "C" void kernel_launch(void* const* d_in, const int* in_sizes, int n_in,
                              void* d_out, int out_size, void* d_ws, size_t ws_size,
                              hipStream_t stream) {
    const float* cls[5] = {(const float*)d_in[0], (const float*)d_in[3],
                           (const float*)d_in[6], (const float*)d_in[9],
                           (const float*)d_in[12]};
    const float* bbx[5] = {(const float*)d_in[1], (const float*)d_in[4],
                           (const float*)d_in[7], (const float*)d_in[10],
                           (const float*)d_in[13]};
    const float* ctr[5] = {(const float*)d_in[2], (const float*)d_in[5],
                           (const float*)d_in[8], (const float*)d_in[11],
                           (const float*)d_in[14]};
    const float* gtb  = (const float*)d_in[15];
    const int*   glab = (const int*)d_in[16];
    float* out = (float*)d_out;

    // workspace layout
    char*  w   = (char*)d_ws;
    float* acc = (float*)w;                        // 8 floats (first 256B slab)
    int*   s1  = (int*)(w + 256);                  // BN ints
    int*   s2  = s1 + BN;                          // BN ints
    float* U   = (float*)(s2 + BN);                // NB*NG*8
    float* V1  = U + NB * NG * 8;                  // NANC16*8
    float* V2  = V1 + NANC16 * 8;                  // NANC16*8
    float* P   = V2 + NANC16 * 8;                  // NANC*2

    k_init<<<512, 256, 0, stream>>>(acc, s1, 2 * BN);
    k_anchor<<<(NANC16 + 255) / 256, 256, 0, stream>>>(V1, V2, P);
    k_gtfeat<<<(NB * NG + 255) / 256, 256, 0, stream>>>(gtb, U);
    k_assign<<<NB, 32, 0, stream>>>(U, V1, V2, s1, s2);
    k_cls<<<(BN / 4 + 255) / 256, 256, 0, stream>>>(cls[0], cls[1], cls[2], cls[3],
                                                    cls[4], s1, s2, glab, acc);
    k_box<<<(BN + 255) / 256, 256, 0, stream>>>(bbx[0], bbx[1], bbx[2], bbx[3],
                                                bbx[4], ctr[0], ctr[1], ctr[2],
                                                ctr[3], ctr[4], gtb, P, s1, s2, acc);
    k_final<<<1, 1, 0, stream>>>(acc, out);
}